// GCN_35450660062089
// MI455X (gfx1250) — compile-verified
//
#include <hip/hip_runtime.h>
#include <hip/hip_bf16.h>

typedef __attribute__((ext_vector_type(2))) float v2f;
typedef __attribute__((ext_vector_type(8))) float v8f;

#define N_NODES 50000

// ---------------------------------------------------------------------------
// GEMM via V_WMMA_F32_16X16X4_F32: T[M,DOUT] = act(H[M,K]) @ Wm[K,DOUT]
// One wave per 16x16 output tile; K steps of 4 chained through the f32 WMMA.
// A layout (16x4 f32): lane&15 = M row, VGPR j holds K = kb + j + 2*(lane>>4)
// B layout (4x16 f32): lane&15 = N col, VGPR j holds K = kb + j + 2*(lane>>4)
// C/D layout: lane&15 = N col, VGPR r holds M = tileM*16 + r + 8*(lane>>4)
// ---------------------------------------------------------------------------
template<int K, int DOUT, bool RELU>
__global__ void gcn_gemm_wmma(const float* __restrict__ H,
                              const float* __restrict__ Wm,
                              float* __restrict__ T, int M) {
  const int NT = DOUT / 16;                       // tiles along N
  int wid  = (int)((blockIdx.x * (unsigned)blockDim.x + threadIdx.x) >> 5);
  int lane = threadIdx.x & 31;
  int tM = wid / NT;
  int tN = wid % NT;
  if (tM * 16 >= M) return;                       // wave-uniform exit (EXEC stays all-1s)

  int sub  = lane & 15;
  int hi   = lane >> 4;                           // 0 or 1 (half-wave)
  int row  = tM * 16 + sub;                       // A-matrix row for this lane
  int col  = tN * 16 + sub;                       // B/C column for this lane
  int koff = hi * 2;

  v8f acc = {};
  #pragma unroll
  for (int kb = 0; kb < K; kb += 4) {
    float a0 = H[row * K + kb + koff + 0];
    float a1 = H[row * K + kb + koff + 1];
    if (RELU) { a0 = fmaxf(a0, 0.0f); a1 = fmaxf(a1, 0.0f); }
    v2f A = { a0, a1 };
    v2f B = { Wm[(kb + koff + 0) * DOUT + col],
              Wm[(kb + koff + 1) * DOUT + col] };
    acc = __builtin_amdgcn_wmma_f32_16x16x4_f32(
        /*neg_a=*/false, A, /*neg_b=*/false, B,
        /*c_mod=*/(short)0, acc, /*reuse_a=*/false, /*reuse_b=*/false);
  }

  int rowBase = tM * 16 + hi * 8;
  #pragma unroll
  for (int r = 0; r < 8; ++r)
    T[(rowBase + r) * DOUT + col] = acc[r];
}

// G[n, :] = bias broadcast (fuses "+b" into the aggregation accumulator init)
template<int DOUT>
__global__ void bias_init(float* __restrict__ G, const float* __restrict__ b,
                          int total) {
  int i = blockIdx.x * blockDim.x + threadIdx.x;
  if (i < total) G[i] = b[i & (DOUT - 1)];
}

// Edge-parallel scatter-add: G[dst[e], :] += T[src[e], :]
// One thread per float4 chunk -> DOUT/4 threads per edge; src/dst index loads
// are uniform across the chunk group (hardware broadcast), data L2-resident.
template<int DOUT>
__global__ void edge_scatter(const float* __restrict__ T,
                             const int* __restrict__ src,
                             const int* __restrict__ dst,
                             float* __restrict__ G, int E) {
  const int CH = DOUT / 4;
  long long gid = (long long)blockIdx.x * blockDim.x + threadIdx.x;
  if (gid >= (long long)E * CH) return;
  int e = (int)(gid / CH);
  int c = (int)(gid % CH);
  int s = src[e];
  int d = dst[e];
  const float4 v = *((const float4*)(T + (size_t)s * DOUT + (size_t)c * 4));
  float* gp = G + (size_t)d * DOUT + (size_t)c * 4;
  unsafeAtomicAdd(gp + 0, v.x);   // -> global_atomic_add_f32 (L2 atomic units)
  unsafeAtomicAdd(gp + 1, v.y);
  unsafeAtomicAdd(gp + 2, v.z);
  unsafeAtomicAdd(gp + 3, v.w);
}

// One wave32 per node, lane == class (32 classes): xor-shuffle max + sum.
__global__ void logsoftmax32(const float* __restrict__ X, float* __restrict__ out,
                             int N) {
  int wid  = (int)((blockIdx.x * (unsigned)blockDim.x + threadIdx.x) >> 5);
  int lane = threadIdx.x & 31;
  if (wid >= N) return;                           // wave-uniform
  float x = X[(size_t)wid * 32 + lane];
  float m = x;
  #pragma unroll
  for (int o = 16; o > 0; o >>= 1) m = fmaxf(m, __shfl_xor(m, o, 32));
  float ex = expf(x - m);
  float s = ex;
  #pragma unroll
  for (int o = 16; o > 0; o >>= 1) s += __shfl_xor(s, o, 32);
  out[(size_t)wid * 32 + lane] = x - m - logf(s);
}

extern "C" void kernel_launch(void* const* d_in, const int* in_sizes, int n_in,
                              void* d_out, int out_size, void* d_ws, size_t ws_size,
                              hipStream_t stream) {
  const float* feat = (const float*)d_in[0];
  const float* W0   = (const float*)d_in[1];
  const float* b0   = (const float*)d_in[2];
  const float* W1   = (const float*)d_in[3];
  const float* b1   = (const float*)d_in[4];
  const float* W2   = (const float*)d_in[5];
  const float* b2   = (const float*)d_in[6];
  const int*   src  = (const int*)d_in[7];
  const int*   dst  = (const int*)d_in[8];
  const int    E    = in_sizes[7];

  // Workspace: T (transformed features) | G (aggregation accumulator)
  float* T = (float*)d_ws;
  float* G = (float*)((char*)d_ws + (size_t)N_NODES * 64 * sizeof(float));
  float* out = (float*)d_out;

  const int BLK = 256;
  auto blocks_for = [](long long threads, int blk) {
    return (unsigned)((threads + blk - 1) / blk);
  };

  // ----- Layer 0: T = feat @ W0 ; G = b0 ; G += scatter(T) -----
  {
    long long waves = (long long)(N_NODES / 16) * (64 / 16);
    gcn_gemm_wmma<64, 64, false>
        <<<blocks_for(waves * 32, BLK), BLK, 0, stream>>>(feat, W0, T, N_NODES);
    int total = N_NODES * 64;
    bias_init<64><<<blocks_for(total, BLK), BLK, 0, stream>>>(G, b0, total);
    edge_scatter<64>
        <<<blocks_for((long long)E * 16, BLK), BLK, 0, stream>>>(T, src, dst, G, E);
  }
  // ----- Layer 1: T = relu(G) @ W1 ; G = b1 ; G += scatter(T) -----
  {
    long long waves = (long long)(N_NODES / 16) * (64 / 16);
    gcn_gemm_wmma<64, 64, true>
        <<<blocks_for(waves * 32, BLK), BLK, 0, stream>>>(G, W1, T, N_NODES);
    int total = N_NODES * 64;
    bias_init<64><<<blocks_for(total, BLK), BLK, 0, stream>>>(G, b1, total);
    edge_scatter<64>
        <<<blocks_for((long long)E * 16, BLK), BLK, 0, stream>>>(T, src, dst, G, E);
  }
  // ----- Layer 2 (32-wide): T = relu(G) @ W2 ; G = b2 ; G += scatter(T) -----
  {
    long long waves = (long long)(N_NODES / 16) * (32 / 16);
    gcn_gemm_wmma<64, 32, true>
        <<<blocks_for(waves * 32, BLK), BLK, 0, stream>>>(G, W2, T, N_NODES);
    int total = N_NODES * 32;
    bias_init<32><<<blocks_for(total, BLK), BLK, 0, stream>>>(G, b2, total);
    edge_scatter<32>
        <<<blocks_for((long long)E * 8, BLK), BLK, 0, stream>>>(T, src, dst, G, E);
  }
  // ----- log_softmax over 32 classes, one wave per node -----
  logsoftmax32<<<blocks_for((long long)N_NODES * 32, BLK), BLK, 0, stream>>>(
      G, out, N_NODES);
}